// NeuralLorenz96_46815143526765
// MI455X (gfx1250) — compile-verified
//
#include <hip/hip_runtime.h>
#include <stdint.h>

// Two-scale Lorenz-96, RK4, sequential over time. Single persistent workgroup,
// state in LDS (triple-buffered), per-thread state in registers, history rows
// streamed out with gfx1250 async store-from-LDS (ASYNCcnt).

#define L96_K    64     // slow vars
#define L96_J    32     // fast per slow
#define L96_NY   2048   // K*J
#define L96_NTH  256    // 8 wave32 waves
#define L96_YPT  8      // Y elements per thread (256*8 = 2048)

__launch_bounds__(L96_NTH, 1)
__global__ void neural_lorenz96_kernel(const float* __restrict__ X0,
                                       const float* __restrict__ Y0,
                                       const float* __restrict__ ncp,
                                       const float* __restrict__ sF,
                                       const float* __restrict__ sh,
                                       const float* __restrict__ sb,
                                       const float* __restrict__ sc,
                                       const float* __restrict__ sdt,
                                       float* __restrict__ xout,
                                       float* __restrict__ yout,
                                       int nt)
{
    // layout of each buffer: [ X(64) | Y(2048) ]
    __shared__ __align__(16) float bufA[L96_K + L96_NY];  // current state
    __shared__ __align__(16) float bufB[L96_K + L96_NY];  // stage state (ping)
    __shared__ __align__(16) float bufC[L96_K + L96_NY];  // stage state (pong)

    const int  t   = (int)threadIdx.x;
    const int  y0i = t * L96_YPT;       // first Y index owned by this thread
    const int  xk  = t >> 2;            // X index coupled to this Y block (8t/32)
    const bool isx = (t < L96_K);       // waves 0-1 also own one X each

    const float F   = *sF;
    const float h   = *sh;
    const float b   = *sb;
    const float c   = *sc;
    const float dt  = *sdt;
    const float cb  = c * b;
    const float hcb = h * c / b;
    const float hdt = 0.5f * dt;
    const float dt6 = dt * (1.0f / 6.0f);

    float ya[L96_YPT];    // current Y state (registers)
    float acc[L96_YPT];   // RK accumulator: k1 + 2k2 + 2k3 (+k4)
    float yo1[L96_YPT];   // stage-state registers (ping)
    float yo2[L96_YPT];   // stage-state registers (pong)
    float xa = 0.f, xo1 = 0.f, xo2 = 0.f, xacc = 0.f;
    float ncv = 0.f;      // neural coupling for current step (row n-1)

    // ---- init: load row 0 of state, emit history row 0 ----
    #pragma unroll
    for (int j = 0; j < L96_YPT; ++j) {
        float v = Y0[y0i + j];
        ya[j] = v;
        bufA[L96_K + y0i + j] = v;
        yout[y0i + j] = v;
    }
    if (isx) {
        xa = X0[t];
        bufA[t] = xa;
        xout[t] = xa;
        ncv = ncp[t];          // nc row 0 (used by step n=1)
    }
    __syncthreads();

    // raw LDS byte address of this thread's Y chunk in bufA (flat-ptr low 32 bits
    // are the LDS offset per CDNA5 aperture rules) — source for async stores.
    const uint32_t ldsAy = (uint32_t)(uintptr_t)(&bufA[L96_K + y0i]);

    // One RK stage: halo from LDS buffer S, own values from registers yin/xin,
    // write stage state (ya + coef*k) to LDS buffer D and registers yonext/xonext,
    // accumulate w*k into acc.
    auto stage = [&](const float* __restrict__ S, float* __restrict__ D,
                     const float* yin, float xin, float coef, float w,
                     float* yonext, float* xonext) {
        float xs  = S[xk];                                            // X coupling
        float hm1 = S[L96_K + ((y0i + L96_NY - 1) & (L96_NY - 1))];   // Y[i0-1]
        float hp0 = S[L96_K + ((y0i + L96_YPT)     & (L96_NY - 1))];  // Y[i0+8]
        float hp1 = S[L96_K + ((y0i + L96_YPT + 1) & (L96_NY - 1))];  // Y[i0+9]
        float v[L96_YPT + 3];
        v[0] = hm1;
        #pragma unroll
        for (int j = 0; j < L96_YPT; ++j) v[j + 1] = yin[j];
        v[L96_YPT + 1] = hp0;
        v[L96_YPT + 2] = hp1;
        const float fx = hcb * xs;
        #pragma unroll
        for (int j = 0; j < L96_YPT; ++j) {
            // Ydot[i] = -c*b*Y[i+1]*(Y[i+2]-Y[i-1]) - c*Y[i] + (h*c/b)*X[i/J]
            float k = fmaf(-cb * v[j + 2], v[j + 3] - v[j], fmaf(-c, v[j + 1], fx));
            acc[j] += w * k;
            float nv = fmaf(coef, k, ya[j]);
            yonext[j] = nv;
            D[L96_K + y0i + j] = nv;
        }
        if (isx) {
            float xm1 = S[(t + L96_K - 1) & (L96_K - 1)];
            float xp1 = S[(t + 1)         & (L96_K - 1)];
            float xm2 = S[(t + L96_K - 2) & (L96_K - 1)];
            // Xdot[k] = X[k-1]*(X[k+1]-X[k-2]) - X[k] + F - c*nc[k]
            float kx = fmaf(xm1, xp1 - xm2, F - xin) - c * ncv;
            xacc += w * kx;
            float nx = fmaf(coef, kx, xa);
            *xonext = nx;
            D[t] = nx;
        }
    };

    for (int n = 1; n < nt; ++n) {
        // prefetch next step's coupling row a full step early (latency hidden)
        float ncnext = isx ? ncp[n * L96_K + t] : 0.f;

        #pragma unroll
        for (int j = 0; j < L96_YPT; ++j) acc[j] = 0.f;
        xacc = 0.f;

        stage(bufA, bufB, ya,  xa,  hdt, 1.0f, yo1, &xo1);  // k1 -> state B
        __syncthreads();
        stage(bufB, bufC, yo1, xo1, hdt, 2.0f, yo2, &xo2);  // k2 -> state C
        __syncthreads();
        stage(bufC, bufB, yo2, xo2, dt,  2.0f, yo1, &xo1);  // k3 -> state B
        __syncthreads();

        // ---- stage 4 + final RK update (writes bufA) ----
        // previous step's async stores read bufA: per-wave wait (a full step of
        // slack has passed, so this is essentially free)
        asm volatile("s_wait_asynccnt 0x0" ::: "memory");
        {
            const float* S = bufB;
            float xs  = S[xk];
            float hm1 = S[L96_K + ((y0i + L96_NY - 1) & (L96_NY - 1))];
            float hp0 = S[L96_K + ((y0i + L96_YPT)     & (L96_NY - 1))];
            float hp1 = S[L96_K + ((y0i + L96_YPT + 1) & (L96_NY - 1))];
            float v[L96_YPT + 3];
            v[0] = hm1;
            #pragma unroll
            for (int j = 0; j < L96_YPT; ++j) v[j + 1] = yo1[j];
            v[L96_YPT + 1] = hp0;
            v[L96_YPT + 2] = hp1;
            const float fx = hcb * xs;
            #pragma unroll
            for (int j = 0; j < L96_YPT; ++j) {
                float k  = fmaf(-cb * v[j + 2], v[j + 3] - v[j], fmaf(-c, v[j + 1], fx));
                float nv = fmaf(dt6, acc[j] + k, ya[j]);   // Y + dt/6*(k1+2k2+2k3+k4)
                ya[j] = nv;
                bufA[L96_K + y0i + j] = nv;
            }
            if (isx) {
                float xm1 = S[(t + L96_K - 1) & (L96_K - 1)];
                float xp1 = S[(t + 1)         & (L96_K - 1)];
                float xm2 = S[(t + L96_K - 2) & (L96_K - 1)];
                float kx  = fmaf(xm1, xp1 - xm2, F - xo1) - c * ncv;
                float nx  = fmaf(dt6, xacc + kx, xa);
                xa = nx;
                bufA[t] = nx;
                xout[(size_t)n * L96_K + t] = nx;          // X history: tiny, plain store
            }
        }
        // make this wave's bufA writes visible before the async engine reads them
        asm volatile("s_wait_dscnt 0x0" ::: "memory");
        {
            // stream this step's 2048-float Y row from LDS to HBM, fire-and-forget
            uint64_t g = (uint64_t)(uintptr_t)(yout + (size_t)n * L96_NY + y0i);
            uint32_t l = ldsAy;
            asm volatile("global_store_async_from_lds_b128 %0, %1, off"
                         :: "v"(g), "v"(l) : "memory");
            asm volatile("global_store_async_from_lds_b128 %0, %1, off offset:16"
                         :: "v"(g), "v"(l) : "memory");
        }
        __syncthreads();
        ncv = ncnext;
    }
    // s_endpgm performs an implicit wait-idle (drains ASYNCcnt/STOREcnt)
}

extern "C" void kernel_launch(void* const* d_in, const int* in_sizes, int n_in,
                              void* d_out, int out_size, void* d_ws, size_t ws_size,
                              hipStream_t stream) {
    (void)n_in; (void)out_size; (void)d_ws; (void)ws_size;
    const float* X0  = (const float*)d_in[0];   // [NT, 64]
    const float* Y0  = (const float*)d_in[1];   // [NT, 2048]
    const float* ncp = (const float*)d_in[2];   // [NT, 64]
    const float* sF  = (const float*)d_in[3];
    const float* sh  = (const float*)d_in[4];
    const float* sb  = (const float*)d_in[5];
    const float* sc  = (const float*)d_in[6];
    const float* sdt = (const float*)d_in[7];

    const int nt = in_sizes[0] / L96_K;         // 20000
    float* xout = (float*)d_out;                          // [NT, 64]
    float* yout = xout + (size_t)nt * L96_K;              // [NT, 2048]

    neural_lorenz96_kernel<<<dim3(1), dim3(L96_NTH), 0, stream>>>(
        X0, Y0, ncp, sF, sh, sb, sc, sdt, xout, yout, nt);
}